// MinibatchDiscrimination_16552803959337
// MI455X (gfx1250) — compile-verified
//
#include <hip/hip_runtime.h>
#include <hip/hip_bf16.h>

typedef __attribute__((ext_vector_type(2))) float v2f;
typedef __attribute__((ext_vector_type(8))) float v8f;

#define B_DIM 512
#define F_DIM 1024
#define N_COL 500   // K*D = 100*5
#define N_PAD 512   // padded column count
#define NUM_K 100
#define DIM_D 5

// ---------------------------------------------------------------------------
// Stage 0: prepack T [1024, 500] row-major -> Tp [512, 1024] column-major
// (Tp[col][k] = T[k][col], zero for col >= 500). One-time 2 MB move; makes
// the GEMM B-operand a contiguous b64 load with no predication.
// ---------------------------------------------------------------------------
__global__ __launch_bounds__(256)
void mbd_pack_T(const float* __restrict__ T, float* __restrict__ Tp) {
    // one thread per element over 512*1024 elements; thread id maps to (col, k) with k fastest
    int idx = blockIdx.x * 256 + threadIdx.x;          // 0 .. 512*1024-1
    const int k   = idx & (F_DIM - 1);
    const int col = idx >> 10;
    float v = 0.0f;
    if (col < N_COL) v = T[(size_t)k * N_COL + col];
    Tp[(size_t)col * F_DIM + k] = v;
}

// ---------------------------------------------------------------------------
// Stage 1: X = inputs @ T via V_WMMA_F32_16X16X4_F32 (exact fp32 path).
// One wave computes one 16x16 tile; 256 K-steps of depth 4, unrolled x4.
// A lane mapping (16x4 f32): row M = lane%16, K pair {kb, kb+1}, kb = (lane>>4)*2
// B lane mapping (4x16 f32): col N = lane%16, same K pair  -> contiguous in Tp
// ---------------------------------------------------------------------------
__global__ __launch_bounds__(32)
void mbd_gemm_wmma(const float* __restrict__ A,    // [512, 1024] row-major
                   const float* __restrict__ Tp,   // [512, 1024] col-major packed
                   float* __restrict__ X) {        // [512, 512] padded
    const int tileM = blockIdx.x * 16;
    const int tileN = blockIdx.y * 16;
    const int lane  = threadIdx.x;      // 0..31
    const int half  = lane >> 4;        // 0 or 1
    const int l16   = lane & 15;
    const int kb    = half * 2;

    const float* aptr = A  + (size_t)(tileM + l16) * F_DIM + kb;   // A[row][kb]
    const float* bptr = Tp + (size_t)(tileN + l16) * F_DIM + kb;   // Tp[col][kb]

    v8f c = {};
    #pragma unroll 4
    for (int kk = 0; kk < F_DIM; kk += 4) {
        const v2f a = *(const v2f*)(aptr + kk);   // A[row][kk+kb .. kk+kb+1]
        const v2f b = *(const v2f*)(bptr + kk);   // B[kk+kb .. kk+kb+1][col]
        // 8 args: (neg_a, A, neg_b, B, c_mod, C, reuse_a, reuse_b)
        c = __builtin_amdgcn_wmma_f32_16x16x4_f32(
                false, a, false, b, (short)0, c, false, false);
    }

    // D layout: VGPR r -> (M = r + 8*half, N = l16)
    #pragma unroll
    for (int r = 0; r < 8; ++r) {
        const int row = tileM + r + half * 8;
        X[(size_t)row * N_PAD + tileN + l16] = c[r];
    }
}

// ---------------------------------------------------------------------------
// Stage 2: out[i,k] = sum_j exp(-sum_d |x[i,k,d]-x[j,k,d]|)
// One block per row i; thread k owns one output. X is L2-resident (1 MB).
// ---------------------------------------------------------------------------
__global__ __launch_bounds__(128)
void mbd_pairwise(const float* __restrict__ X,   // [512, 512] padded
                  float* __restrict__ out) {     // [512, 100]
    const int i = blockIdx.x;
    const int k = threadIdx.x;
    if (k >= NUM_K) return;

    float xi[DIM_D];
    #pragma unroll
    for (int d = 0; d < DIM_D; ++d)
        xi[d] = X[(size_t)i * N_PAD + k * DIM_D + d];

    float acc = 0.0f;
    #pragma unroll 4
    for (int j = 0; j < B_DIM; ++j) {
        const float* xj = X + (size_t)j * N_PAD + k * DIM_D;
        float l1 = fabsf(xi[0] - xj[0]);
        l1 += fabsf(xi[1] - xj[1]);
        l1 += fabsf(xi[2] - xj[2]);
        l1 += fabsf(xi[3] - xj[3]);
        l1 += fabsf(xi[4] - xj[4]);
        acc += __expf(-l1);             // v_exp_f32 path
    }
    out[(size_t)i * NUM_K + k] = acc;
}

// ---------------------------------------------------------------------------
extern "C" void kernel_launch(void* const* d_in, const int* in_sizes, int n_in,
                              void* d_out, int out_size, void* d_ws, size_t ws_size,
                              hipStream_t stream) {
    const float* inputs = (const float*)d_in[0];   // [512, 1024] fp32
    const float* T      = (const float*)d_in[1];   // [1024, 500] fp32
    float*       out    = (float*)d_out;           // [512, 100]  fp32

    float* X  = (float*)d_ws;                      // [512, 512]  (1 MB)
    float* Tp = X + (size_t)B_DIM * N_PAD;         // [512, 1024] (2 MB)

    (void)in_sizes; (void)n_in; (void)out_size; (void)ws_size;

    mbd_pack_T<<<dim3((N_PAD * F_DIM) / 256), dim3(256), 0, stream>>>(T, Tp);

    dim3 gemmGrid(B_DIM / 16, N_PAD / 16);         // 32 x 32 tiles
    mbd_gemm_wmma<<<gemmGrid, dim3(32), 0, stream>>>(inputs, Tp, X);

    mbd_pairwise<<<dim3(B_DIM), dim3(128), 0, stream>>>(X, out);
}